// MolecularPropertyPrediction_51290499449389
// MI455X (gfx1250) — compile-verified
//
#include <hip/hip_runtime.h>

typedef __attribute__((ext_vector_type(2))) float v2f;
typedef __attribute__((ext_vector_type(8))) float v8f;

#define DTOT   16384   // total atoms
#define NA     64      // atoms per molecule
#define NDIM   64      // feature dim
#define NLAYER 3
#define LROW   68      // padded LDS row stride (floats): bank = (4*row + col) % 64

__global__ __launch_bounds__(256)
void gnn_mol_wmma_kernel(const int*   __restrict__ atoms,
                         const float* __restrict__ adj,     // [16384,16384], block-diagonal
                         const float* __restrict__ embed,   // [N_FP,64]
                         const float* __restrict__ Wl,      // [3,64,64]
                         const float* __restrict__ bl,      // [3,64]
                         const float* __restrict__ Wp,      // [64]
                         const float* __restrict__ bp,      // [1]
                         float*       __restrict__ out)     // [256]
{
    __shared__ float xs [NA * LROW];   // molecule feature state x        [atom][dim]
    __shared__ float hsT[NA * LROW];   // per-layer message h, TRANSPOSED [dim][atom]
    __shared__ float As [NA * LROW];   // 64x64 diagonal adjacency block  [atom][atom]
    __shared__ float Ws [NA * LROW];   // current layer weights           [out][in]
    __shared__ float red[256];

    const int b    = blockIdx.x;      // molecule id
    const int t    = threadIdx.x;
    const int lane = t & 31;          // wave32
    const int wid  = t >> 5;          // 8 waves

    const int r  = t >> 2;            // 0..63  (cooperative-copy row)
    const int c0 = (t & 3) * 16;      // 0,16,32,48

    // ---- stage x = embed_table[atoms] and the ONLY nonzero adjacency block ----
    {
        const int fp = atoms[b * NA + r];
        const float* esrc = embed + (size_t)fp * NDIM + c0;
        const float* asrc = adj + (size_t)(b * NA + r) * DTOT + (size_t)b * NA + c0;
        float* xdst = xs + r * LROW + c0;
        float* adst = As + r * LROW + c0;
        #pragma unroll
        for (int i = 0; i < 4; ++i) {
            *(float4*)(xdst + i * 4) = *(const float4*)(esrc + i * 4);
            *(float4*)(adst + i * 4) = *(const float4*)(asrc + i * 4);
        }
    }

    // WMMA f32 16x16x4 fragment coordinates (ISA 7.12.2 layouts):
    //  A[m,k]:   vgpr = k&1, lane = (k>>1)*16 + m
    //  B[k,n]:   vgpr = k&1, lane = (k>>1)*16 + n
    //  C/D[m,n]: vgpr = m&7, lane = (m>>3)*16 + n
    const int fr   = lane & 15;          // tile-local m (A) / n (B,C,D)
    const int koff = (lane >> 4) * 2;    // which K pair this half-wave holds
    const int moff = (lane >> 4) * 8;    // C/D row offset for upper half-wave

    for (int l = 0; l < NLAYER; ++l) {
        // ---- stage W_l into LDS (coalesced, once per layer) ----
        {
            const float* wsrc = Wl + l * NDIM * NDIM + r * NDIM + c0;
            float* wdst = Ws + r * LROW + c0;
            #pragma unroll
            for (int i = 0; i < 4; ++i)
                *(float4*)(wdst + i * 4) = *(const float4*)(wsrc + i * 4);
        }
        const float* bias = bl + l * NDIM;
        __syncthreads();   // Ws ready; x ready (initial load or previous layer)

        // ---- GEMM1: h = relu(x @ W^T + bias); B[k,n] = W[n,k] = Ws[n][k] ----
        #pragma unroll
        for (int ti = 0; ti < 2; ++ti) {
            const int tile = wid * 2 + ti;          // 0..15
            const int tm = (tile >> 2) * 16;
            const int tn = (tile & 3) * 16;
            v8f acc = {};
            #pragma unroll
            for (int kk = 0; kk < NDIM; kk += 4) {
                const int k0 = kk + koff;
                v2f a = *(const v2f*)(xs + (tm + fr) * LROW + k0);
                v2f w = *(const v2f*)(Ws + (tn + fr) * LROW + k0);
                acc = __builtin_amdgcn_wmma_f32_16x16x4_f32(
                          false, a, false, w, (short)0, acc, false, false);
            }
            const float bn = bias[tn + fr];
            #pragma unroll
            for (int v = 0; v < 8; ++v) {
                float h = acc[v] + bn;
                h = h > 0.0f ? h : 0.0f;
                hsT[(tn + fr) * LROW + tm + moff + v] = h;   // transposed store
            }
        }
        __syncthreads();   // h complete; also fences GEMM1's reads of xs

        // ---- GEMM2: x = x + A_blk @ h  (C preloaded with x tile) ----
        //      B[k=atom, n=dim] = h[atom,dim] = hsT[n][k]  -> b64 frag load
        #pragma unroll
        for (int ti = 0; ti < 2; ++ti) {
            const int tile = wid * 2 + ti;
            const int tm = (tile >> 2) * 16;
            const int tn = (tile & 3) * 16;
            v8f acc;
            #pragma unroll
            for (int v = 0; v < 8; ++v)
                acc[v] = xs[(tm + moff + v) * LROW + tn + fr];
            #pragma unroll
            for (int kk = 0; kk < NA; kk += 4) {
                const int k0 = kk + koff;
                v2f a = *(const v2f*)(As  + (tm + fr) * LROW + k0);
                v2f h = *(const v2f*)(hsT + (tn + fr) * LROW + k0);
                acc = __builtin_amdgcn_wmma_f32_16x16x4_f32(
                          false, a, false, h, (short)0, acc, false, false);
            }
            #pragma unroll
            for (int v = 0; v < 8; ++v)
                xs[(tm + moff + v) * LROW + tn + fr] = acc[v];
        }
        __syncthreads();   // x updated for next layer
    }

    // ---- fused pooling + property head: z = sum_{i,d} x[i,d]*Wp[d] + bp ----
    {
        float s = 0.0f;
        #pragma unroll
        for (int c = 0; c < 16; ++c)
            s += xs[r * LROW + c0 + c] * Wp[c0 + c];
        red[t] = s;
    }
    __syncthreads();
    if (t < 32) {
        float s = 0.0f;
        #pragma unroll
        for (int i = 0; i < 8; ++i) s += red[t + i * 32];
        #pragma unroll
        for (int off = 16; off > 0; off >>= 1)
            s += __shfl_down(s, off, 32);
        if (t == 0) out[b] = s + bp[0];
    }
}

extern "C" void kernel_launch(void* const* d_in, const int* in_sizes, int n_in,
                              void* d_out, int out_size, void* d_ws, size_t ws_size,
                              hipStream_t stream) {
    const int*   atoms = (const int*)  d_in[0];
    const float* adj   = (const float*)d_in[1];
    // d_in[2] = seg_ids (unused: block structure is implicit, 64 atoms/molecule)
    const float* embed = (const float*)d_in[3];
    const float* Wl    = (const float*)d_in[4];
    const float* bl    = (const float*)d_in[5];
    const float* Wp    = (const float*)d_in[6];
    const float* bp    = (const float*)d_in[7];
    float* out = (float*)d_out;

    gnn_mol_wmma_kernel<<<256, 256, 0, stream>>>(atoms, adj, embed, Wl, bl, Wp, bp, out);
}